// NonLocalBlock_6519760355924
// MI455X (gfx1250) — compile-verified
//
#include <hip/hip_runtime.h>

#define B_  16
#define C_  256
#define IC_ 32
#define M_  16384

typedef float v2f __attribute__((ext_vector_type(2)));
typedef float v8f __attribute__((ext_vector_type(8)));
typedef unsigned int v4u __attribute__((ext_vector_type(4)));
typedef int v4i __attribute__((ext_vector_type(4)));
typedef int v8i __attribute__((ext_vector_type(8)));

__device__ __forceinline__ v8f wmma_f32(v2f a, v2f b, v8f c) {
  // D = A(16x4 f32) * B(4x16 f32) + C(16x16 f32)
  return __builtin_amdgcn_wmma_f32_16x16x4_f32(
      /*neg_a=*/false, a, /*neg_b=*/false, b,
      /*c_mod=*/(short)0, c, /*reuse_a=*/false, /*reuse_b=*/false);
}

// ---------------------------------------------------------------------------
// Tensor Data Mover: 2D tile (tile_d1 rows x tile_d0 elems, 4B elements) from
// global to LDS with hardware row padding. Issued by one wave; tracked with
// TENSORcnt.
// ---------------------------------------------------------------------------
#if defined(__has_builtin)
#if __has_builtin(__builtin_amdgcn_tensor_load_to_lds) && \
    __has_builtin(__builtin_amdgcn_s_wait_tensorcnt)
#define USE_TDM 1
#endif
#endif
#ifndef USE_TDM
#define USE_TDM 0
#endif

#if USE_TDM
__device__ __forceinline__ void tdm_load_tile_2d(
    unsigned int lds_off, const float* gptr,
    unsigned int tile_d0, unsigned int tile_d1,
    unsigned int tens_d0, unsigned int tens_d1, unsigned int stride0,
    unsigned int pad_int_code, unsigned int pad_amt_code) {
  unsigned long long ga = (unsigned long long)(size_t)(const void*)gptr;
  v4u g0;
  g0[0] = 1u;                                   // count=1, user mode, no gather
  g0[1] = lds_off;                              // lds_addr (bytes)
  g0[2] = (unsigned int)ga;                     // global_addr[31:0]
  g0[3] = (unsigned int)((ga >> 32) & 0x01FFFFFFull) | (2u << 30); // type=2
  v8i g1;
  g1[0] = (int)((2u << 16)                      // data_size = 4B
              | (1u << 20)                      // pad_enable
              | (pad_int_code << 22)            // pad interval (2^(code+1) DW)
              | (pad_amt_code << 25));          // pad amount (code+1 DW)
  g1[1] = (int)((tens_d0 & 0xFFFFu) << 16);     // tensor_dim0[15:0]
  g1[2] = (int)(((tens_d0 >> 16) & 0xFFFFu) | ((tens_d1 & 0xFFFFu) << 16));
  g1[3] = (int)(((tens_d1 >> 16) & 0xFFFFu) | ((tile_d0 & 0xFFFFu) << 16));
  g1[4] = (int)(tile_d1 & 0xFFFFu);             // tile_dim1 (tile_dim2 = 0)
  g1[5] = (int)stride0;                         // tensor_dim0_stride[31:0]
  g1[6] = 0;                                    // stride0 hi / dim1_stride
  g1[7] = 0;
  v4i gz = {0, 0, 0, 0};
#if __clang_major__ >= 23
  v8i gz8 = {0, 0, 0, 0, 0, 0, 0, 0};
  __builtin_amdgcn_tensor_load_to_lds(g0, g1, gz, gz, gz8, 0);
#else
  __builtin_amdgcn_tensor_load_to_lds(g0, g1, gz, gz, 0);
#endif
}
#endif

__device__ __forceinline__ unsigned int lds_offset_of(const void* p) {
  return (unsigned int)(size_t)p;   // LDS aperture: addr[31:0] = LDS offset
}

// ---------------------------------------------------------------------------
// Kernel 1: per-batch Gram XX[b] += x_b x_b^T and channel sums, via WMMA f32.
// Grid (M/MCH1, B), block 256 (8 waves). Wave w owns i-tiles 2w, 2w+1 (32 ch)
// so every B-frag feeds two WMMAs. x staged by TDM, double-buffered.
// ---------------------------------------------------------------------------
#define KS1  16
#define KSP1 18     // padded LDS row stride (floats)
#define MCH1 1024
#define NS1  (MCH1 / KS1)   // 64 K-steps per workgroup

__global__ __launch_bounds__(256) void nlb_gram_kernel(
    const float* __restrict__ x, float* __restrict__ XX,
    float* __restrict__ xsum) {
  __shared__ float st[2][C_ * KSP1];    // 2 x 18 KB
  __shared__ float ssum[8][2][32];

  const int b    = blockIdx.y;
  const int tid  = threadIdx.x;
  const int wave = tid >> 5;
  const int lane = tid & 31;
  const int lh   = lane & 15;
  const int kh   = (lane >> 4) * 2;
  const int rhi  = (lane >= 16) ? 8 : 0;

  const float* xb = x + (size_t)b * C_ * M_;
  const int m_begin = blockIdx.x * MCH1;

  v8f acc0[16], acc1[16];
  v8f z = {};
  #pragma unroll
  for (int j = 0; j < 16; ++j) { acc0[j] = z; acc1[j] = z; }
  float cs0 = 0.0f, cs1 = 0.0f;

#if USE_TDM
  if (wave == 0)
    tdm_load_tile_2d(lds_offset_of(&st[0][0]), xb + m_begin,
                     KS1, C_, M_, C_, M_, 3u, 1u);
#endif
  for (int s = 0; s < NS1; ++s) {
    const float* buf = st[s & 1];
    __syncthreads();                    // all waves done with buf[(s-1)&1]
#if USE_TDM
    if (wave == 0) {
      if (s + 1 < NS1) {
        tdm_load_tile_2d(lds_offset_of(&st[(s + 1) & 1][0]),
                         xb + m_begin + (s + 1) * KS1,
                         KS1, C_, M_, C_, M_, 3u, 1u);
        __builtin_amdgcn_s_wait_tensorcnt(1);   // oldest (tile s) landed
      } else {
        __builtin_amdgcn_s_wait_tensorcnt(0);
      }
    }
#else
    {
      const int m0 = m_begin + s * KS1;
      float* dstb = (float*)st[s & 1];
      #pragma unroll
      for (int i = 0; i < 4; ++i) {
        int f  = tid + i * 256;         // float4 index in [0, 1024)
        int ch = f >> 2;
        int k4 = (f & 3) * 4;
        float4 v = *(const float4*)(xb + (size_t)ch * M_ + m0 + k4);
        __builtin_prefetch(xb + (size_t)ch * M_ + m0 + KS1 + k4, 0, 1);
        float* d = &dstb[ch * KSP1 + k4];
        d[0] = v.x; d[1] = v.y; d[2] = v.z; d[3] = v.w;
      }
    }
#endif
    __syncthreads();                    // buf[s&1] published
    #pragma unroll
    for (int kf = 0; kf < KS1 / 4; ++kf) {
      const int kk = kf * 4 + kh;
      v2f a0 = *(const v2f*)&buf[(wave * 32 + lh) * KSP1 + kk];
      v2f a1 = *(const v2f*)&buf[(wave * 32 + 16 + lh) * KSP1 + kk];
      cs0 += a0.x + a0.y;
      cs1 += a1.x + a1.y;
      #pragma unroll
      for (int j = 0; j < 16; ++j) {
        v2f bf = *(const v2f*)&buf[(j * 16 + lh) * KSP1 + kk];
        acc0[j] = wmma_f32(a0, bf, acc0[j]);
        acc1[j] = wmma_f32(a1, bf, acc1[j]);
      }
    }
  }

  // channel partial sums: lanes l / l+16 of a frag pair cover all K offsets
  ssum[wave][0][lane] = cs0;
  ssum[wave][1][lane] = cs1;
  __syncthreads();
  if (tid < C_) {                       // channel tid = 32w + 16it + l
    int w = tid >> 5, it = (tid >> 4) & 1, l = tid & 15;
    atomicAdd(&xsum[b * C_ + tid], ssum[w][it][l] + ssum[w][it][l + 16]);
  }

  float* XXb = XX + (size_t)b * C_ * C_;
  #pragma unroll
  for (int j = 0; j < 16; ++j) {
    #pragma unroll
    for (int r = 0; r < 8; ++r) {
      int cj = j * 16 + lh;
      atomicAdd(&XXb[(wave * 32 + r + rhi) * C_ + cj],      acc0[j][r]);
      atomicAdd(&XXb[(wave * 32 + 16 + r + rhi) * C_ + cj], acc1[j][r]);
    }
  }
}

// ---------------------------------------------------------------------------
// Kernel 2: per-batch attention + effective weights (tiny; scalar fp32).
//   Cov = XX/M - xbar xbar^T ; sigma = theta Cov theta^T / sqrt(IC)
//   attn = softmax(sigma) ; W_eff = w_w attn g_w ; b_eff = w_w attn g_b + w_b
// ---------------------------------------------------------------------------
__global__ __launch_bounds__(256) void nlb_attn_kernel(
    const float* __restrict__ XX, const float* __restrict__ xsum,
    const float* __restrict__ theta_w, const float* __restrict__ g_w,
    const float* __restrict__ g_b, const float* __restrict__ w_w,
    const float* __restrict__ w_b, float* __restrict__ Weff,
    float* __restrict__ beff) {
  __shared__ float P[IC_][C_ + 1];
  __shared__ float xbar[C_];
  __shared__ float sig[IC_][IC_ + 1];
  __shared__ float q[IC_];
  __shared__ float bg[IC_];

  const int b = blockIdx.x;
  const int t = threadIdx.x;
  const float* XXb = XX + (size_t)b * C_ * C_;
  const float inv_m = 1.0f / (float)M_;

  xbar[t] = xsum[b * C_ + t] * inv_m;
  __syncthreads();

  {
    float p[IC_];
    #pragma unroll
    for (int i = 0; i < IC_; ++i) p[i] = 0.0f;
    for (int d = 0; d < C_; ++d) {
      float xv = XXb[d * C_ + t];
      #pragma unroll
      for (int i = 0; i < IC_; ++i) p[i] += theta_w[i * C_ + d] * xv;
    }
    #pragma unroll
    for (int i = 0; i < IC_; ++i) P[i][t] = p[i];
  }
  if (t < IC_) {
    float s = 0.0f;
    for (int c = 0; c < C_; ++c) s += theta_w[t * C_ + c] * xbar[c];
    q[t] = s;
  }
  __syncthreads();

  const float scl = rsqrtf((float)IC_);
  for (int e = t; e < IC_ * IC_; e += 256) {
    int i = e >> 5, j = e & 31;
    float s = 0.0f;
    for (int c = 0; c < C_; ++c) s += P[i][c] * theta_w[j * C_ + c];
    sig[i][j] = (s * inv_m - q[i] * q[j]) * scl;
  }
  __syncthreads();

  if (t < IC_) {
    float mx = -1e30f;
    for (int j = 0; j < IC_; ++j) mx = fmaxf(mx, sig[t][j]);
    float sum = 0.0f;
    for (int j = 0; j < IC_; ++j) { float e = __expf(sig[t][j] - mx); sig[t][j] = e; sum += e; }
    float inv = 1.0f / sum;
    float s = 0.0f;
    for (int j = 0; j < IC_; ++j) { sig[t][j] *= inv; s += sig[t][j] * g_b[j]; }
    bg[t] = s;
  }
  __syncthreads();

  {
    float gcol[IC_], a2[IC_];
    #pragma unroll
    for (int j = 0; j < IC_; ++j) gcol[j] = g_w[j * C_ + t];
    #pragma unroll
    for (int i = 0; i < IC_; ++i) {
      float s = 0.0f;
      #pragma unroll
      for (int j = 0; j < IC_; ++j) s += sig[i][j] * gcol[j];
      a2[i] = s;
    }
    float* Wb = Weff + (size_t)b * C_ * C_;
    for (int co = 0; co < C_; ++co) {
      float s = 0.0f;
      #pragma unroll
      for (int i = 0; i < IC_; ++i) s += w_w[co * IC_ + i] * a2[i];
      Wb[co * C_ + t] = s;
    }
  }
  {
    float s = w_b[t];
    #pragma unroll
    for (int i = 0; i < IC_; ++i) s += w_w[t * IC_ + i] * bg[i];
    beff[b * C_ + t] = s;
  }
}

// ---------------------------------------------------------------------------
// Kernel 3: out = W_eff[b] @ x_b + x_b + b_eff. Grid (M/MCH3, B), block 256
// (8 waves). Wave w keeps TWO 16-row W_eff slabs resident (co = 32w..32w+31)
// as v2f A-frags; x streamed through LDS (TDM, double-buffered) as B-frags,
// each feeding two WMMAs.
// ---------------------------------------------------------------------------
#define MT3  16
#define MP3  17     // padded LDS row stride (floats)
#define MCH3 256
#define NS3  (MCH3 / MT3)   // 16 m-tiles per workgroup

__global__ __launch_bounds__(256) void nlb_apply_kernel(
    const float* __restrict__ x, const float* __restrict__ Weff,
    const float* __restrict__ beff, float* __restrict__ out) {
  __shared__ float st[2][C_ * MP3];     // 2 x 17 KB

  const int b    = blockIdx.y;
  const int tid  = threadIdx.x;
  const int wave = tid >> 5;
  const int lane = tid & 31;
  const int lh   = lane & 15;
  const int kh   = (lane >> 4) * 2;
  const int rhi  = (lane >= 16) ? 8 : 0;

  const float* xb = x + (size_t)b * C_ * M_;
  float*       ob = out + (size_t)b * C_ * M_;
  const float* Wb = Weff + (size_t)b * C_ * C_;
  const int co0 = wave * 32;

  // resident A-frags: lane l holds W[co + (l&15)][4kf + kh + {0,1}]
  v2f wf0[64], wf1[64];
  #pragma unroll
  for (int kf = 0; kf < 64; ++kf) {
    wf0[kf] = *(const v2f*)&Wb[(co0 + lh) * C_ + kf * 4 + kh];
    wf1[kf] = *(const v2f*)&Wb[(co0 + 16 + lh) * C_ + kf * 4 + kh];
  }
  float be0[8], be1[8];
  #pragma unroll
  for (int r = 0; r < 8; ++r) {
    be0[r] = beff[b * C_ + co0 + r + rhi];
    be1[r] = beff[b * C_ + co0 + 16 + r + rhi];
  }

  const int m_begin = blockIdx.x * MCH3;
#if USE_TDM
  if (wave == 0)
    tdm_load_tile_2d(lds_offset_of(&st[0][0]), xb + m_begin,
                     MT3, C_, M_, C_, M_, 3u, 0u);
#endif
  for (int s = 0; s < NS3; ++s) {
    const float* buf = st[s & 1];
    const int m0 = m_begin + s * MT3;
    __syncthreads();
#if USE_TDM
    if (wave == 0) {
      if (s + 1 < NS3) {
        tdm_load_tile_2d(lds_offset_of(&st[(s + 1) & 1][0]),
                         xb + m_begin + (s + 1) * MT3,
                         MT3, C_, M_, C_, M_, 3u, 0u);
        __builtin_amdgcn_s_wait_tensorcnt(1);
      } else {
        __builtin_amdgcn_s_wait_tensorcnt(0);
      }
    }
#else
    {
      float* dstb = (float*)st[s & 1];
      #pragma unroll
      for (int i = 0; i < 4; ++i) {
        int f  = tid + i * 256;
        int ch = f >> 2;
        int k4 = (f & 3) * 4;
        float4 v = *(const float4*)(xb + (size_t)ch * M_ + m0 + k4);
        __builtin_prefetch(xb + (size_t)ch * M_ + m0 + MT3 + k4, 0, 1);
        float* d = &dstb[ch * MP3 + k4];
        d[0] = v.x; d[1] = v.y; d[2] = v.z; d[3] = v.w;
      }
    }
#endif
    __syncthreads();

    v8f acc0 = {}, acc1 = {};
    #pragma unroll
    for (int kf = 0; kf < 64; ++kf) {
      int ci = kf * 4 + kh;             // B-frag: VGPR0 -> K=kh, VGPR1 -> kh+1
      v2f bf;
      bf.x = buf[ci * MP3 + lh];
      bf.y = buf[(ci + 1) * MP3 + lh];
      acc0 = wmma_f32(wf0[kf], bf, acc0);
      acc1 = wmma_f32(wf1[kf], bf, acc1);
    }
    // fused epilogue: + residual + bias (64B-contiguous per half-wave)
    #pragma unroll
    for (int r = 0; r < 8; ++r) {
      size_t i0 = (size_t)(co0 + r + rhi) * M_ + m0 + lh;
      size_t i1 = (size_t)(co0 + 16 + r + rhi) * M_ + m0 + lh;
      ob[i0] = acc0[r] + xb[i0] + be0[r];
      ob[i1] = acc1[r] + xb[i1] + be1[r];
    }
  }
}

// ---------------------------------------------------------------------------
extern "C" void kernel_launch(void* const* d_in, const int* in_sizes, int n_in,
                              void* d_out, int out_size, void* d_ws, size_t ws_size,
                              hipStream_t stream) {
  const float* x       = (const float*)d_in[0];
  const float* g_w     = (const float*)d_in[1];
  const float* g_b     = (const float*)d_in[2];
  const float* theta_w = (const float*)d_in[3];
  // d_in[4] = theta_b: cancels exactly in the covariance -> unused
  const float* w_w     = (const float*)d_in[5];
  const float* w_b     = (const float*)d_in[6];
  float* out = (float*)d_out;

  char* ws = (char*)d_ws;
  const size_t XX_BYTES  = (size_t)B_ * C_ * C_ * sizeof(float);   // 4 MB
  const size_t SUM_BYTES = (size_t)B_ * C_ * sizeof(float);
  float* XX   = (float*)ws;
  float* xsum = (float*)(ws + XX_BYTES);
  float* Weff = (float*)(ws + XX_BYTES + SUM_BYTES);
  float* beff = (float*)(ws + XX_BYTES + SUM_BYTES + XX_BYTES);

  hipMemsetAsync(ws, 0, XX_BYTES + SUM_BYTES, stream);   // zero accumulators

  nlb_gram_kernel<<<dim3(M_ / MCH1, B_), 256, 0, stream>>>(x, XX, xsum);

  nlb_attn_kernel<<<B_, 256, 0, stream>>>(XX, xsum, theta_w, g_w, g_b, w_w, w_b,
                                          Weff, beff);

  nlb_apply_kernel<<<dim3(M_ / MCH3, B_), 256, 0, stream>>>(x, Weff, beff, out);
}